// GraphGCN_5222680232660
// MI455X (gfx1250) — compile-verified
//
#include <hip/hip_runtime.h>
#include <hip/hip_bf16.h>

typedef __attribute__((ext_vector_type(2))) float v2f;
typedef __attribute__((ext_vector_type(8))) float v8f;

#if defined(__gfx1250__) && __has_builtin(__builtin_amdgcn_wmma_f32_16x16x4_f32)
#define USE_WMMA_F32 1
#else
#define USE_WMMA_F32 0
#endif

#define N_NODES 50000
#define N_EDGES 800000
#define N_GRAPHS 64
#define HID 64

// ---------------- small utility kernels ----------------

__global__ void fill_k(float* p, float v, int n) {
    int i = blockIdx.x * blockDim.x + threadIdx.x;
    if (i < n) p[i] = v;
}

__global__ void deg_k(const long long* __restrict__ ei, float* __restrict__ deg, int nE) {
    int e = blockIdx.x * blockDim.x + threadIdx.x;
    if (e < nE) atomicAdd(&deg[ei[nE + e]], 1.0f);   // dst = ei[1][e]
}

__global__ void rsqrt_k(float* __restrict__ d, int n) {
    int i = blockIdx.x * blockDim.x + threadIdx.x;
    if (i < n) d[i] = rsqrtf(d[i]);
}

__global__ void enorm_k(const long long* __restrict__ ei, const float* __restrict__ dinv,
                        float* __restrict__ enorm, int nE) {
    int e = blockIdx.x * blockDim.x + threadIdx.x;
    if (e < nE) enorm[e] = dinv[ei[e]] * dinv[ei[nE + e]];
}

// ---------------- fp32 WMMA GEMM: C[M,N] = A[M,K] * B[K,N] ----------------
// One wave per 16x16 output tile; K stepped by 4 via V_WMMA_F32_16X16X4_F32.
// M must be a multiple of 16 (50000 = 3125*16), N a multiple of 16 (64).

__global__ __launch_bounds__(32)
void gemm_wmma_f32(const float* __restrict__ A, const float* __restrict__ B,
                   float* __restrict__ C, int K, int N, int ntiles) {
    int tile = blockIdx.x;
    int mt = tile / ntiles;
    int nt = tile % ntiles;
    int lane = threadIdx.x;
    int lo = lane & 15;
    int hi = lane >> 4;                       // 0 or 1

    const float* Arow = A + (size_t)(mt * 16 + lo) * K;   // A row for this lane
    const float* Bcol = B + (nt * 16 + lo);               // B column for this lane
    v8f acc = {};

#if USE_WMMA_F32
    for (int k = 0; k < K; k += 4) {
        int kb = k + 2 * hi;                  // lanes 0-15: K=k,k+1 ; lanes 16-31: K=k+2,k+3
        v2f a, b;
        a.x = Arow[kb];
        a.y = Arow[kb + 1];
        b.x = Bcol[(size_t)kb * N];
        b.y = Bcol[(size_t)(kb + 1) * N];
        acc = __builtin_amdgcn_wmma_f32_16x16x4_f32(
            /*neg_a=*/false, a, /*neg_b=*/false, b,
            /*c_mod=*/(short)0, acc, /*reuse_a=*/false, /*reuse_b=*/false);
    }
#else
    // scalar fallback with identical output mapping
    const float* Ab = A + (size_t)(mt * 16 + 8 * hi) * K;
    for (int k = 0; k < K; ++k) {
        float bv = Bcol[(size_t)k * N];
        (void)Arow;
        #pragma unroll
        for (int v = 0; v < 8; ++v) acc[v] += Ab[(size_t)v * K + k] * bv;
    }
#endif

    // C/D layout: VGPR v -> row M = v + 8*hi, col N = lane&15
    float* Cb = C + (size_t)(mt * 16 + 8 * hi) * N + nt * 16 + lo;
    #pragma unroll
    for (int v = 0; v < 8; ++v) Cb[(size_t)v * N] = acc[v];
}

// ---------------- GCN layer pieces ----------------

// agg[i][c] = dinv[i]^2 * hw[i][c] + bias[c]   (self-loop term + bias)
__global__ void init_agg_k(const float* __restrict__ hw, const float* __restrict__ dinv,
                           const float* __restrict__ bias, float* __restrict__ agg, int n) {
    int t = blockIdx.x * blockDim.x + threadIdx.x;
    int i = t >> 6, c = t & 63;
    if (i < n) {
        float di = dinv[i];
        agg[t] = di * di * hw[t] + bias[c];
    }
}

// agg[dst][c] += hw[src][c] * enorm[e]  -- 64 consecutive lanes share an edge
__global__ void edge_agg_k(const long long* __restrict__ ei, const float* __restrict__ enorm,
                           const float* __restrict__ hw, float* __restrict__ agg, int nE) {
    int t = blockIdx.x * blockDim.x + threadIdx.x;
    int e = t >> 6, c = t & 63;
    if (e < nE) {
        long long s = ei[e];
        long long d = ei[nE + e];
        atomicAdd(&agg[(size_t)d * HID + c], hw[(size_t)s * HID + c] * enorm[e]);
    }
}

// h = relu(agg / max(||agg||_2, eps)) per node; one wave per node (64 feats)
__global__ __launch_bounds__(256)
void norm_relu_k(const float* __restrict__ agg, float* __restrict__ h, int n) {
    int wave = (blockIdx.x * blockDim.x + threadIdx.x) >> 5;
    int lane = threadIdx.x & 31;
    if (wave >= n) return;
    const float* a = agg + (size_t)wave * HID;
    float x0 = a[lane];
    float x1 = a[lane + 32];
    float ss = x0 * x0 + x1 * x1;
    #pragma unroll
    for (int o = 16; o > 0; o >>= 1) ss += __shfl_xor(ss, o, 32);
    float inv = 1.0f / fmaxf(sqrtf(ss), 1e-12f);
    float* ho = h + (size_t)wave * HID;
    ho[lane]      = fmaxf(x0 * inv, 0.0f);
    ho[lane + 32] = fmaxf(x1 * inv, 0.0f);
}

// ---------------- pooling + output ----------------

__global__ void count_k(const long long* __restrict__ batch, float* __restrict__ cnt, int n) {
    int i = blockIdx.x * blockDim.x + threadIdx.x;
    if (i < n) atomicAdd(&cnt[batch[i]], 1.0f);
}

// pooled layout: [G][128]: cols 0..63 = running sum, cols 64..127 = running max (uint-bit max, h>=0)
__global__ void pool_k(const float* __restrict__ h, const long long* __restrict__ batch,
                       float* __restrict__ pooled, int n) {
    int t = blockIdx.x * blockDim.x + threadIdx.x;
    int i = t >> 6, c = t & 63;
    if (i < n) {
        long long g = batch[i];
        float v = h[t];
        atomicAdd(&pooled[(size_t)g * 128 + c], v);
        atomicMax(reinterpret_cast<unsigned int*>(&pooled[(size_t)g * 128 + 64 + c]),
                  __float_as_uint(v));
    }
}

__global__ void fin_mean_k(float* __restrict__ pooled, const float* __restrict__ cnt) {
    int t = blockIdx.x * blockDim.x + threadIdx.x;   // G*64 threads
    int g = t >> 6, c = t & 63;
    if (g < N_GRAPHS) pooled[(size_t)g * 128 + c] /= fmaxf(cnt[g], 1.0f);
}

// out[g][j] = sum_k pooled[g][k] * Wl[k][j] + bl[j]   (64x32 out, K=128; tiny)
__global__ void linear_k(const float* __restrict__ pooled, const float* __restrict__ Wl,
                         const float* __restrict__ bl, float* __restrict__ out) {
    int t = blockIdx.x * blockDim.x + threadIdx.x;   // G*32 threads
    int g = t >> 5, j = t & 31;
    if (g >= N_GRAPHS) return;
    float s = bl[j];
    const float* p = pooled + (size_t)g * 128;
    #pragma unroll 8
    for (int k = 0; k < 128; ++k) s += p[k] * Wl[(size_t)k * 32 + j];
    out[t] = s;
}

// ---------------- driver ----------------

extern "C" void kernel_launch(void* const* d_in, const int* in_sizes, int n_in,
                              void* d_out, int out_size, void* d_ws, size_t ws_size,
                              hipStream_t stream) {
    const float*     x     = (const float*)d_in[0];
    const long long* ei    = (const long long*)d_in[1];   // [2, E] int64
    const long long* batch = (const long long*)d_in[2];   // [N] int64
    const float* W1 = (const float*)d_in[3];
    const float* b1 = (const float*)d_in[4];
    const float* W2 = (const float*)d_in[5];
    const float* b2 = (const float*)d_in[6];
    const float* W3 = (const float*)d_in[7];
    const float* b3 = (const float*)d_in[8];
    const float* Wl = (const float*)d_in[9];
    const float* bl = (const float*)d_in[10];
    float* out = (float*)d_out;

    const int N = N_NODES, E = N_EDGES, G = N_GRAPHS;

    float* ws     = (float*)d_ws;
    float* dinv   = ws;                          // N (holds deg first)
    float* enorm  = dinv + 50048;                // E
    float* hw     = enorm + E;                   // N*64
    float* agg    = hw  + (size_t)N * HID;       // N*64
    float* h      = agg + (size_t)N * HID;       // N*64
    float* pooled = h   + (size_t)N * HID;       // G*128
    float* cnt    = pooled + (size_t)G * 128;    // G   (contiguous after pooled)

    const int mt = N / 16;          // 3125, exact
    const int nt = HID / 16;        // 4
    const int NC = N * HID;         // 3.2M

    // degrees (init 1.0 for self-loop), dinv, edge norms
    fill_k <<<(N + 255) / 256, 256, 0, stream>>>(dinv, 1.0f, N);
    deg_k  <<<(E + 255) / 256, 256, 0, stream>>>(ei, dinv, E);
    rsqrt_k<<<(N + 255) / 256, 256, 0, stream>>>(dinv, N);
    enorm_k<<<(E + 255) / 256, 256, 0, stream>>>(ei, dinv, enorm, E);

    // layer 1: K=128
    gemm_wmma_f32<<<mt * nt, 32, 0, stream>>>(x, W1, hw, 128, HID, nt);
    init_agg_k <<<NC / 256, 256, 0, stream>>>(hw, dinv, b1, agg, N);
    edge_agg_k <<<(E * 64) / 256, 256, 0, stream>>>(ei, enorm, hw, agg, E);
    norm_relu_k<<<(N * 32) / 256 + 1, 256, 0, stream>>>(agg, h, N);

    // layer 2: K=64
    gemm_wmma_f32<<<mt * nt, 32, 0, stream>>>(h, W2, hw, HID, HID, nt);
    init_agg_k <<<NC / 256, 256, 0, stream>>>(hw, dinv, b2, agg, N);
    edge_agg_k <<<(E * 64) / 256, 256, 0, stream>>>(ei, enorm, hw, agg, E);
    norm_relu_k<<<(N * 32) / 256 + 1, 256, 0, stream>>>(agg, h, N);

    // layer 3: K=64
    gemm_wmma_f32<<<mt * nt, 32, 0, stream>>>(h, W3, hw, HID, HID, nt);
    init_agg_k <<<NC / 256, 256, 0, stream>>>(hw, dinv, b3, agg, N);
    edge_agg_k <<<(E * 64) / 256, 256, 0, stream>>>(ei, enorm, hw, agg, E);
    norm_relu_k<<<(N * 32) / 256 + 1, 256, 0, stream>>>(agg, h, N);

    // pooling: zero pooled+cnt (contiguous G*128 + G), then accumulate
    fill_k <<<(G * 128 + G + 255) / 256, 256, 0, stream>>>(pooled, 0.0f, G * 128 + G);
    count_k<<<(N + 255) / 256, 256, 0, stream>>>(batch, cnt, N);
    pool_k <<<(N * 64) / 256, 256, 0, stream>>>(h, batch, pooled, N);
    fin_mean_k<<<(G * 64 + 255) / 256, 256, 0, stream>>>(pooled, cnt);

    // final linear -> out [G,32]
    linear_k<<<(G * 32 + 255) / 256, 256, 0, stream>>>(pooled, Wl, bl, out);
}